// EquivariantLayer_34437047779346
// MI455X (gfx1250) — compile-verified
//
#include <hip/hip_runtime.h>

// ---------------------------------------------------------------------------
// Problem constants (from the reference)
// ---------------------------------------------------------------------------
#define C_IN   8
#define C_OUT  16
#define R_IN   4
#define R_OUT  4
#define P_DIM  5
#define T_DIM  16
#define B_DIM  80            // P*T
#define N_DIM  20000
#define KTOT   2560          // B * C_IN * R_IN
#define NCOLS  64            // C_OUT * R_OUT
#define MTILE  16            // output rows per workgroup

typedef float v2f __attribute__((ext_vector_type(2)));
typedef float v8f __attribute__((ext_vector_type(8)));

// ---------------------------------------------------------------------------
// Kernel 1: rebuild ROT (deterministic integer recurrence) and gather the
// permuted weight matrix into a WMMA-B-fragment-swizzled layout in d_ws.
//
// lwB flat layout: [kt (0..639)][ct (0..3)][lane (0..31)][i2 (0..1)] where
//   col  = ct*16 + (lane & 15)
//   krow = kt*4 + (lane>>4)*2 + i2
// so each wave's per-iteration B-fragment load is one coalesced float2.
// ---------------------------------------------------------------------------
__global__ void build_lw_kernel(const float* __restrict__ weights,
                                float* __restrict__ lwB) {
    __shared__ int rot[R_IN * B_DIM * R_OUT];   // small[s][j], s = ri*80 + b

    if (threadIdx.x == 0) {
        int pif[R_IN * B_DIM];
        int tmp[R_IN * B_DIM];
        for (int i = 0; i < R_IN * B_DIM; ++i) {
            pif[i] = i;
            rot[i * R_OUT + 0] = i;
        }
        for (int j = 1; j < R_OUT; ++j) {
            // _angle_rotation: view as (64 rows x 5); rotate each block of 16
            // rows up by one.
            for (int i = 0; i < R_IN; ++i) {
                for (int r = 0; r < T_DIM - 1; ++r)
                    for (int cc = 0; cc < P_DIM; ++cc)
                        tmp[(i * T_DIM + r) * P_DIM + cc] =
                            pif[(i * T_DIM + r + 1) * P_DIM + cc];
                for (int cc = 0; cc < P_DIM; ++cc)
                    tmp[(i * T_DIM + T_DIM - 1) * P_DIM + cc] =
                        pif[(i * T_DIM) * P_DIM + cc];
            }
            // _kernel_rotation: view as (4 x 80); rotate rows down by one.
            for (int blk = 0; blk < R_IN; ++blk) {
                int src = (blk + R_IN - 1) & 3;
                for (int cc = 0; cc < T_DIM * P_DIM; ++cc)
                    pif[blk * T_DIM * P_DIM + cc] = tmp[src * T_DIM * P_DIM + cc];
            }
            for (int i = 0; i < R_IN * B_DIM; ++i) rot[i * R_OUT + j] = pif[i];
        }
    }
    __syncthreads();

    // lw2[krow][col] = weights[ci*320 + rot[(ri*80 + b)*4 + ro]][co]
    //   krow = b*32 + ci*4 + ri ; col = co*4 + ro
    for (int idx = threadIdx.x; idx < KTOT * NCOLS; idx += blockDim.x) {
        int i2   = idx & 1;
        int lane = (idx >> 1) & 31;
        int ctk  = idx >> 6;           // kt*4 + ct
        int ct   = ctk & 3;
        int kt   = ctk >> 2;
        int col  = ct * 16 + (lane & 15);
        int krow = kt * 4 + (lane >> 4) * 2 + i2;

        int b  = krow >> 5;
        int f  = krow & 31;
        int ci = f >> 2;
        int ri = f & 3;
        int co = col >> 2;
        int ro = col & 3;
        int w  = ci * (R_IN * B_DIM) + rot[(ri * B_DIM + b) * R_OUT + ro];
        lwB[idx] = weights[w * C_OUT + co];
    }
}

// ---------------------------------------------------------------------------
// Kernel 2: fused gather/segment-sum (in LDS) + WMMA GEMM.
// One workgroup (8 waves) owns 16 output rows. The 16x2560 f32 agg tile
// lives entirely in LDS (160 KB < 320 KB/WGP) so it never touches HBM.
// k addressing is XOR-swizzled by (m<<2) so the per-lane A-fragment
// ds_load_b64s hit 16 distinct banks without padding.
// ---------------------------------------------------------------------------
__global__ __launch_bounds__(256)
void fused_equiv_kernel(const float* __restrict__ x,
                        const int*   __restrict__ rows,
                        const int*   __restrict__ cols,
                        const float* __restrict__ vals,
                        const float* __restrict__ lwB,
                        float*       __restrict__ out,
                        int nnz) {
    __shared__ float agg[MTILE * KTOT];    // 163,840 B
    __shared__ float stage[4 * 256];       // K-half reduction staging
    __shared__ int   seg[2];

    const int tid = threadIdx.x;
    const int n0  = blockIdx.x * MTILE;
    const int r0  = n0 * B_DIM;                    // first segment row
    const int r1  = r0 + MTILE * B_DIM;            // one past last

    // ---- zero the agg tile (b128 stores) ----
    for (int i = tid * 4; i < MTILE * KTOT; i += 256 * 4)
        *reinterpret_cast<float4*>(&agg[i]) = make_float4(0.f, 0.f, 0.f, 0.f);

    // ---- binary search sorted conn_rows for this block's edge range ----
    if (tid < 2) {
        int target = (tid == 0) ? r0 : r1;
        int lo = 0, hi = nnz;
        while (lo < hi) {
            int mid = (lo + hi) >> 1;
            if (rows[mid] < target) lo = mid + 1; else hi = mid;
        }
        seg[tid] = lo;
    }
    __syncthreads();

    // ---- edge-parallel scatter: agg[m][b*32+f] += x[c][f] * v ----
    const int e0 = seg[0], e1 = seg[1];
    for (int e = e0 + tid; e < e1; e += 256) {
        int   r = rows[e] - r0;          // (m)*80 + b
        int   m = r / B_DIM;
        int   b = r - m * B_DIM;
        int   c = cols[e];
        float v = vals[e];
        const float4* xr = reinterpret_cast<const float4*>(x + (size_t)c * 32);
        float* dst = &agg[m * KTOT];
        int kbase = b * 32;
        int sw    = (m & 15) << 2;       // XOR swizzle (bits >= 2: float4-safe)
#pragma unroll
        for (int j = 0; j < 8; ++j) {
            float4 xv = xr[j];
            int ks = (kbase + j * 4) ^ sw;
            atomicAdd(&dst[ks + 0], xv.x * v);   // ds_add_f32 (no return)
            atomicAdd(&dst[ks + 1], xv.y * v);
            atomicAdd(&dst[ks + 2], xv.z * v);
            atomicAdd(&dst[ks + 3], xv.w * v);
        }
    }
    __syncthreads();

    // ---- WMMA GEMM: (16 x 2560) @ (2560 x 64), f32 in / f32 accum ----
    // 8 waves: ct = column tile (16 cols each), kh = K half (1280 each).
    const int lane = tid & 31;
    const int wave = tid >> 5;
    const int ct   = wave & 3;
    const int kh   = wave >> 2;

    // A-fragment addressing (ISA 32-bit 16x4 layout):
    //   lanes 0-15 : M=lane,    K = k0+0, k0+1
    //   lanes 16-31: M=lane-16, K = k0+2, k0+3
    const int m    = lane & 15;
    const int koff = (lane >> 4) * 2;
    const int sw   = m << 2;
    const float* aggm = &agg[m * KTOT];
    const v2f* bptr = reinterpret_cast<const v2f*>(lwB);

    v8f acc = {};
    const int kt_beg = kh * 320;
    const int kt_end = kt_beg + 320;
    for (int kt = kt_beg; kt < kt_end; ++kt) {
        int k0 = kt * 4;
        v2f afrag = *reinterpret_cast<const v2f*>(&aggm[(k0 + koff) ^ sw]);
        v2f bfrag = bptr[(kt * 4 + ct) * 32 + lane];
        acc = __builtin_amdgcn_wmma_f32_16x16x4_f32(
                  false, afrag, false, bfrag, (short)0, acc, false, false);
    }

    // ---- reduce the two K-halves through LDS, then store C ----
    if (kh == 1) {
#pragma unroll
        for (int r = 0; r < 8; ++r) stage[ct * 256 + r * 32 + lane] = acc[r];
    }
    __syncthreads();
    if (kh == 0) {
#pragma unroll
        for (int r = 0; r < 8; ++r) acc[r] += stage[ct * 256 + r * 32 + lane];
        // C layout: VGPR r -> row n0 + r + (lane>>4)*8, col ct*16 + (lane&15)
        const int mm = (lane >> 4) * 8;
        const int nn = ct * 16 + (lane & 15);
#pragma unroll
        for (int r = 0; r < 8; ++r)
            out[(size_t)(n0 + r + mm) * NCOLS + nn] = acc[r];
    }
}

// ---------------------------------------------------------------------------
// Host launcher
// ---------------------------------------------------------------------------
extern "C" void kernel_launch(void* const* d_in, const int* in_sizes, int n_in,
                              void* d_out, int out_size, void* d_ws, size_t ws_size,
                              hipStream_t stream) {
    const float* x       = (const float*)d_in[0];
    const int*   rows    = (const int*)  d_in[1];
    const int*   cols    = (const int*)  d_in[2];
    const float* vals    = (const float*)d_in[3];
    const float* weights = (const float*)d_in[4];
    float*       out     = (float*)d_out;
    float*       lwB     = (float*)d_ws;      // 2560*64*4 = 640 KB scratch
    const int    nnz     = in_sizes[1];

    hipLaunchKernelGGL(build_lw_kernel, dim3(1), dim3(256), 0, stream,
                       weights, lwB);
    hipLaunchKernelGGL(fused_equiv_kernel, dim3(N_DIM / MTILE), dim3(256), 0,
                       stream, x, rows, cols, vals, lwB, out, nnz);
}